// GINNet_12567074308657
// MI455X (gfx1250) — compile-verified
//
#include <hip/hip_runtime.h>
#include <hip/hip_bf16.h>

typedef float v2f __attribute__((ext_vector_type(2)));
typedef float v8f __attribute__((ext_vector_type(8)));

#define NNODE_F 602

// ---------------------------------------------------------------------------
// utility kernels
// ---------------------------------------------------------------------------
__global__ void k_zero(float* __restrict__ p, long n) {
  long i = (long)blockIdx.x * blockDim.x + threadIdx.x;
  long st = (long)gridDim.x * blockDim.x;
  for (; i < n; i += st) p[i] = 0.0f;
}

// ---------------------------------------------------------------------------
// h1 = x @ W + b   (x: [n,602] f32, W: [602,64], h1: [n,64])
// one wave per 16-row tile, 4 WMMA accumulators (16x64), K chunked via LDS
// ---------------------------------------------------------------------------
__global__ __launch_bounds__(256) void k_lin1(const float* __restrict__ x,
                                              const float* __restrict__ w,
                                              const float* __restrict__ b,
                                              float* __restrict__ h1, int n) {
  constexpr int F_ = NNODE_F;
  constexpr int FP = 604;     // F padded to multiple of 4
  constexpr int CK = 128;     // K-rows of W staged per chunk (32KB)
  __shared__ float sW[CK * 64];

  const int t = threadIdx.x;
  const int wave = t >> 5, lane = t & 31;
  const int half = lane >> 4, lr = lane & 15;
  const int m0 = (blockIdx.x * 8 + wave) * 16;
  const int arow = m0 + lr;
  const bool av = arow < n;
  const float* xrow = x + (size_t)arow * F_;

  v8f acc[4];
#pragma unroll
  for (int tt = 0; tt < 4; ++tt) acc[tt] = {};

  for (int c0 = 0; c0 < FP; c0 += CK) {
    __syncthreads();
    // stage W rows [c0, c0+CK) into LDS (zero-pad rows >= 602)
    for (int i = t; i < CK * 16; i += 256) {   // float4 granularity
      int r = c0 + (i >> 4);
      float4 v = make_float4(0.f, 0.f, 0.f, 0.f);
      if (r < F_) v = ((const float4*)(w + (size_t)r * 64))[i & 15];
      ((float4*)sW)[i] = v;
    }
    __syncthreads();

    const int kmax = min(CK, FP - c0);
    for (int k0 = 0; k0 < kmax; k0 += 4) {
      const int ka = c0 + k0 + 2 * half;      // global K (even)
      v2f a;
      if (av && ka < F_) {                    // ka even; ka<602 => ka+1<602
        a.x = xrow[ka];
        a.y = xrow[ka + 1];
      } else {
        a.x = 0.0f; a.y = 0.0f;
      }
      const int kl = k0 + 2 * half;           // local K within LDS chunk
#pragma unroll
      for (int tt = 0; tt < 4; ++tt) {
        v2f bf;
        bf.x = sW[kl * 64 + tt * 16 + lr];
        bf.y = sW[(kl + 1) * 64 + tt * 16 + lr];
        acc[tt] = __builtin_amdgcn_wmma_f32_16x16x4_f32(
            false, a, false, bf, (short)0, acc[tt], false, false);
      }
    }
  }

#pragma unroll
  for (int tt = 0; tt < 4; ++tt)
#pragma unroll
    for (int r = 0; r < 8; ++r) {
      int row = m0 + r + 8 * half;
      int col = tt * 16 + lr;
      if (row < n) h1[(size_t)row * 64 + col] = acc[tt][r] + b[col];
    }
}

// ---------------------------------------------------------------------------
// scatter-add:  g[dst[e]][:] += h[src[e]][:]   (DD = 64 or 32)
// ---------------------------------------------------------------------------
template <int DD>
__global__ __launch_bounds__(256) void k_scatter(const float* __restrict__ h,
                                                 const int* __restrict__ src,
                                                 const int* __restrict__ dst,
                                                 float* __restrict__ g, int ne) {
  constexpr int GP = DD / 4;
  const long total = (long)ne * GP;
  const long stride = (long)gridDim.x * 256;
  for (long i = (long)blockIdx.x * 256 + threadIdx.x; i < total; i += stride) {
    const int e = (int)(i / GP);
    const int c0 = (int)(i % GP) * 4;
    const int s = src[e], d = dst[e];
    const float4 v = *(const float4*)(h + (size_t)s * DD + c0);
    float* gp = g + (size_t)d * DD + c0;
    unsafeAtomicAdd(gp + 0, v.x);
    unsafeAtomicAdd(gp + 1, v.y);
    unsafeAtomicAdd(gp + 2, v.z);
    unsafeAtomicAdd(gp + 3, v.w);
  }
}

// ---------------------------------------------------------------------------
// out = relu( in @ W1 + b1 ) @ W2 + b2
//   in = A (+ G if PREADD), A: [n,K1], W1: [K1,32], W2: [32,NOUT]
// one wave per 16-row tile; intermediate 16x32 through per-wave LDS
// ---------------------------------------------------------------------------
template <int K1, int NOUT, bool PREADD>
__global__ __launch_bounds__(256) void k_mlp(const float* __restrict__ A,
                                             const float* __restrict__ G,
                                             const float* __restrict__ W1,
                                             const float* __restrict__ B1,
                                             const float* __restrict__ W2,
                                             const float* __restrict__ B2,
                                             float* __restrict__ out, int n) {
  constexpr int NP2 = ((NOUT + 15) / 16) * 16;
  constexpr int NT2 = NP2 / 16;
  __shared__ float sW1[K1 * 32];
  __shared__ float sW2[32 * NP2];
  __shared__ float sB1[32];
  __shared__ float sB2[NP2];
  __shared__ float sH[8][16 * 33];

  const int t = threadIdx.x;
  for (int i = t; i < K1 * 32; i += 256) sW1[i] = W1[i];
  for (int i = t; i < 32 * NP2; i += 256) {
    int r = i / NP2, c = i % NP2;
    sW2[i] = (c < NOUT) ? W2[r * NOUT + c] : 0.0f;
  }
  if (t < 32) sB1[t] = B1[t];
  if (t < NP2) sB2[t] = (t < NOUT) ? B2[t] : 0.0f;
  __syncthreads();

  const int wave = t >> 5, lane = t & 31;
  const int half = lane >> 4, lr = lane & 15;
  const int m0 = (blockIdx.x * 8 + wave) * 16;
  const int arow = m0 + lr;
  const bool av = arow < n;

  // ---- layer 1: [16,K1] @ [K1,32] ----
  v8f acc1[2];
  acc1[0] = {}; acc1[1] = {};
  for (int k0 = 0; k0 < K1; k0 += 4) {
    const int ka = k0 + 2 * half;
    v2f a;
    if (av) {
      float x0 = A[(size_t)arow * K1 + ka];
      float x1 = A[(size_t)arow * K1 + ka + 1];
      if (PREADD) {
        x0 += G[(size_t)arow * K1 + ka];
        x1 += G[(size_t)arow * K1 + ka + 1];
      }
      a.x = x0; a.y = x1;
    } else {
      a.x = 0.0f; a.y = 0.0f;
    }
#pragma unroll
    for (int tt = 0; tt < 2; ++tt) {
      v2f bf;
      bf.x = sW1[ka * 32 + tt * 16 + lr];
      bf.y = sW1[(ka + 1) * 32 + tt * 16 + lr];
      acc1[tt] = __builtin_amdgcn_wmma_f32_16x16x4_f32(
          false, a, false, bf, (short)0, acc1[tt], false, false);
    }
  }

  // bias + relu -> per-wave LDS tile (C-layout to A-layout conversion)
  float* sh = sH[wave];
#pragma unroll
  for (int tt = 0; tt < 2; ++tt)
#pragma unroll
    for (int r = 0; r < 8; ++r) {
      int row = r + 8 * half, col = tt * 16 + lr;
      float v = acc1[tt][r] + sB1[col];
      sh[row * 33 + col] = v > 0.0f ? v : 0.0f;
    }

  // ---- layer 2: [16,32] @ [32,NP2] ----  (same-wave DS dependency)
  v8f acc2[NT2];
#pragma unroll
  for (int tt = 0; tt < NT2; ++tt) acc2[tt] = {};
  for (int k0 = 0; k0 < 32; k0 += 4) {
    const int ka = k0 + 2 * half;
    v2f a;
    a.x = sh[lr * 33 + ka];
    a.y = sh[lr * 33 + ka + 1];
#pragma unroll
    for (int tt = 0; tt < NT2; ++tt) {
      v2f bf;
      bf.x = sW2[ka * NP2 + tt * 16 + lr];
      bf.y = sW2[(ka + 1) * NP2 + tt * 16 + lr];
      acc2[tt] = __builtin_amdgcn_wmma_f32_16x16x4_f32(
          false, a, false, bf, (short)0, acc2[tt], false, false);
    }
  }

#pragma unroll
  for (int tt = 0; tt < NT2; ++tt)
#pragma unroll
    for (int r = 0; r < 8; ++r) {
      int row = m0 + r + 8 * half;
      int col = tt * 16 + lr;
      if (row < n && col < NOUT)
        out[(size_t)row * NOUT + col] = acc2[tt][r] + sB2[col];
    }
}

// ---------------------------------------------------------------------------
// BatchNorm: column sums / sumsq -> scale/shift -> apply  (32 columns)
// ---------------------------------------------------------------------------
__global__ __launch_bounds__(256) void k_bn_stats(const float* __restrict__ x,
                                                  float* __restrict__ sums,
                                                  int n, int rowsPerBlock) {
  __shared__ float sS[256];
  __shared__ float sQ[256];
  const int t = threadIdx.x;
  const int c = t & 31, grp = t >> 5;
  const int r0 = blockIdx.x * rowsPerBlock;
  const int r1 = min(n, r0 + rowsPerBlock);
  float s = 0.0f, q = 0.0f;
  for (int r = r0 + grp; r < r1; r += 8) {
    float v = x[(size_t)r * 32 + c];
    s += v; q += v * v;
  }
  sS[t] = s; sQ[t] = q;
  __syncthreads();
  if (grp == 0) {
    for (int j = 1; j < 8; ++j) { s += sS[c + 32 * j]; q += sQ[c + 32 * j]; }
    unsafeAtomicAdd(&sums[c], s);
    unsafeAtomicAdd(&sums[32 + c], q);
  }
}

__global__ void k_bn_final(const float* __restrict__ sums,
                           const float* __restrict__ gamma,
                           const float* __restrict__ beta,
                           float* __restrict__ scsh, float invn) {
  int c = threadIdx.x;  // 32 threads
  float mu = sums[c] * invn;
  float var = sums[32 + c] * invn - mu * mu;
  float sc = gamma[c] * rsqrtf(var + 1e-5f);
  scsh[c] = sc;
  scsh[32 + c] = beta[c] - mu * sc;
}

__global__ __launch_bounds__(256) void k_bn_apply(const float* __restrict__ x,
                                                  const float* __restrict__ scsh,
                                                  float* __restrict__ y, long n32) {
  long i = (long)blockIdx.x * 256 + threadIdx.x;
  if (i < n32) {
    int c = (int)(i & 31);
    y[i] = x[i] * scsh[c] + scsh[32 + c];
  }
}

// ---------------------------------------------------------------------------
extern "C" void kernel_launch(void* const* d_in, const int* in_sizes, int n_in,
                              void* d_out, int out_size, void* d_ws, size_t ws_size,
                              hipStream_t stream) {
  const float* x      = (const float*)d_in[0];
  const int*   ei     = (const int*)d_in[1];
  const float* lin1_w = (const float*)d_in[2];
  const float* lin1_b = (const float*)d_in[3];
  const float* nn1_w1 = (const float*)d_in[4];
  const float* nn1_b1 = (const float*)d_in[5];
  const float* nn1_w2 = (const float*)d_in[6];
  const float* nn1_b2 = (const float*)d_in[7];
  const float* bn1_g  = (const float*)d_in[8];
  const float* bn1_b  = (const float*)d_in[9];
  const float* nn2_w1 = (const float*)d_in[10];
  const float* nn2_b1 = (const float*)d_in[11];
  const float* nn2_w2 = (const float*)d_in[12];
  const float* nn2_b2 = (const float*)d_in[13];
  const float* bn2_g  = (const float*)d_in[14];
  const float* bn2_b  = (const float*)d_in[15];
  const float* fc1_w  = (const float*)d_in[16];
  const float* fc1_b  = (const float*)d_in[17];
  const float* fc2_w  = (const float*)d_in[18];
  const float* fc2_b  = (const float*)d_in[19];

  const int n  = in_sizes[0] / NNODE_F;   // 100000
  const int E_ = in_sizes[1] / 2;         // 1600000
  const int* src = ei;
  const int* dst = ei + E_;

  float* ws = (float*)d_ws;
  float* h1 = ws;                           // [n,64]
  float* g1 = ws + (size_t)n * 64;          // [n,64]
  float* t1 = ws + (size_t)n * 128;         // [n,32]
  float* h2 = ws + (size_t)n * 160;         // [n,32]
  float* st = ws + (size_t)n * 192;         // 256 floats of stats
  float* g2 = ws;                           // reuse h1 region [n,32]
  float* h3 = ws + (size_t)n * 32;          // reuse h1 region upper half
  float* t2 = g1;                           // reuse g1 region [n,32]

  const int tileBlocks = (n + 127) / 128;   // 8 waves x 16 rows per block
  const float invn = 1.0f / (float)n;

  // init accumulators
  k_zero<<<512, 256, 0, stream>>>(g1, (long)n * 64);
  k_zero<<<1, 256, 0, stream>>>(st, 256);

  // lin1
  k_lin1<<<tileBlocks, 256, 0, stream>>>(x, lin1_w, lin1_b, h1, n);

  // conv1: aggregate + MLP + BN
  k_scatter<64><<<2048, 256, 0, stream>>>(h1, src, dst, g1, E_);
  k_mlp<64, 32, true><<<tileBlocks, 256, 0, stream>>>(h1, g1, nn1_w1, nn1_b1,
                                                      nn1_w2, nn1_b2, t1, n);
  k_bn_stats<<<(n + 1023) / 1024, 256, 0, stream>>>(t1, st, n, 1024);
  k_bn_final<<<1, 32, 0, stream>>>(st, bn1_g, bn1_b, st + 64, invn);
  k_bn_apply<<<(int)(((long)n * 32 + 255) / 256), 256, 0, stream>>>(t1, st + 64, h2,
                                                                    (long)n * 32);

  // conv2
  k_zero<<<512, 256, 0, stream>>>(g2, (long)n * 32);
  k_scatter<32><<<2048, 256, 0, stream>>>(h2, src, dst, g2, E_);
  k_mlp<32, 32, true><<<tileBlocks, 256, 0, stream>>>(h2, g2, nn2_w1, nn2_b1,
                                                      nn2_w2, nn2_b2, t2, n);
  k_bn_stats<<<(n + 1023) / 1024, 256, 0, stream>>>(t2, st + 128, n, 1024);
  k_bn_final<<<1, 32, 0, stream>>>(st + 128, bn2_g, bn2_b, st + 192, invn);
  k_bn_apply<<<(int)(((long)n * 32 + 255) / 256), 256, 0, stream>>>(t2, st + 192, h3,
                                                                    (long)n * 32);

  // heads: relu(h3 @ fc1 + b) @ fc2 + b -> [n,41]
  k_mlp<32, 41, false><<<tileBlocks, 256, 0, stream>>>(h3, nullptr, fc1_w, fc1_b,
                                                       fc2_w, fc2_b, (float*)d_out, n);
}